// StateCheckpointBank_369367187862
// MI455X (gfx1250) — compile-verified
//
#include <hip/hip_runtime.h>

typedef __attribute__((ext_vector_type(2))) float v2f;
typedef __attribute__((ext_vector_type(8))) float v8f;

#define BB 32
#define TT 4096
#define DD 1024
#define EE 128
#define KK 16
#define CIN (DD + EE)   // 1152
#define KEV 8
#define MAXTIME 512

// ---------------- Kernel 1: per-batch checkpoint selection ----------------
__global__ __launch_bounds__(256) void sel_kernel(const float* __restrict__ scores,
                                                  int* __restrict__ chosen,
                                                  float* __restrict__ out_valid,
                                                  float* __restrict__ out_times) {
    const int b = blockIdx.x;
    const int t = threadIdx.x;
    __shared__ float sval[256];
    __shared__ int   sidx[256];
    __shared__ int   ev[KEV];
    const float* sc = scores + (size_t)b * TT;

    for (int pass = 0; pass < KEV; ++pass) {
        float best = -__builtin_inff();
        int   bidx = TT;  // sentinel (larger than any index)
        for (int i = t; i < TT; i += 256) {
            bool taken = false;
            for (int j = 0; j < pass; ++j) taken |= (ev[j] == i);
            if (!taken) {
                float v = sc[i];
                if (v > best || (v == best && i < bidx)) { best = v; bidx = i; }
            }
        }
        sval[t] = best; sidx[t] = bidx;
        __syncthreads();
        for (int s = 128; s > 0; s >>= 1) {
            if (t < s) {
                float v2 = sval[t + s]; int i2 = sidx[t + s];
                if (v2 > sval[t] || (v2 == sval[t] && i2 < sidx[t])) { sval[t] = v2; sidx[t] = i2; }
            }
            __syncthreads();
        }
        if (t == 0) ev[pass] = sidx[0];
        __syncthreads();
    }

    if (t == 0) {
        int cat[KEV + KK];
        for (int j = 0; j < KEV; ++j) cat[j] = ev[j];
        for (int j = 0; j < KK; ++j)  cat[KEV + j] = j * 273;  // linspace(0,4095,16)
        // insertion sort (24 elements)
        for (int i = 1; i < KEV + KK; ++i) {
            int key = cat[i]; int j = i - 1;
            while (j >= 0 && cat[j] > key) { cat[j + 1] = cat[j]; --j; }
            cat[j + 1] = key;
        }
        // unique, keep 16 smallest (always >= 16 unique -> valid all true)
        int out[KK]; int nu = 0; int prev = -1;
        for (int i = 0; i < KEV + KK && nu < KK; ++i)
            if (cat[i] != prev) { out[nu++] = cat[i]; prev = cat[i]; }
        for (int k = 0; k < KK; ++k) {
            chosen[b * KK + k]    = out[k];
            out_times[b * KK + k] = (float)out[k];
            out_valid[b * KK + k] = 1.0f;
        }
    }
}

// ---------------- Kernel 2: gather state rows + softmax(holder logits) ----------------
__global__ __launch_bounds__(256) void build_raw_kernel(const float* __restrict__ state,
                                                        const float* __restrict__ hlog,
                                                        const int* __restrict__ chosen,
                                                        float* __restrict__ raw,
                                                        float* __restrict__ out_holder) {
    const int k = blockIdx.x, b = blockIdx.y, t = threadIdx.x;
    const int idx = chosen[b * KK + k];
    const float* srow = state + ((size_t)b * TT + idx) * DD;
    float* rrow = raw + ((size_t)b * KK + k) * CIN;
    for (int i = t; i < DD; i += 256) rrow[i] = srow[i];

    __shared__ float sh[256];
    const float* hrow = hlog + ((size_t)b * TT + idx) * EE;
    float v = (t < EE) ? hrow[t] : -__builtin_inff();
    if (t < EE) out_holder[((size_t)b * KK + k) * EE + t] = v;   // raw logits (valid==1)
    sh[t] = v; __syncthreads();
    for (int s = 128; s > 0; s >>= 1) { if (t < s) sh[t] = fmaxf(sh[t], sh[t + s]); __syncthreads(); }
    float m = sh[0]; __syncthreads();
    float e = (t < EE) ? expf(v - m) : 0.0f;
    sh[t] = e; __syncthreads();
    for (int s = 128; s > 0; s >>= 1) { if (t < s) sh[t] += sh[t + s]; __syncthreads(); }
    float sum = sh[0];
    if (t < EE) rrow[DD + t] = e / sum;
}

// ---------------- Kernel 3: f32 WMMA GEMM  proj = raw @ W^T + b  (+ time_table) ----------------
// One wave per 16(M=checkpoints) x 16(N=output-dim) tile; K=1152 in steps of 4.
// A (16x4 f32) lane layout: M = lane&15, v0/v1 hold K = 2*(lane>>4) + {0,1}.
// B (4x16 f32) lane layout: N = lane&15, v0/v1 hold K = 2*(lane>>4) + {0,1}.
// C/D (16x16 f32): 8 VGPRs; lane<16 holds rows 0..7 (col=lane), lane>=16 rows 8..15.
__global__ __launch_bounds__(256) void gemm_wmma_kernel(const float* __restrict__ raw,
                                                        const float* __restrict__ W,
                                                        const float* __restrict__ bias,
                                                        const float* __restrict__ ttab,
                                                        const int* __restrict__ chosen,
                                                        float* __restrict__ entries) {
    const int lane = threadIdx.x & 31;
    const int wave = threadIdx.x >> 5;
    const int b    = blockIdx.y;
    const int o0   = (blockIdx.x * 8 + wave) * 16;   // N-tile start in output dim
    const int n    = lane & 15;
    const int half = lane >> 4;                      // selects K-pair {0,1} vs {2,3}

    const float* Aptr = raw + ((size_t)b * KK + (lane & 15)) * CIN + half * 2;
    const float* Bptr = W + (size_t)(o0 + n) * CIN + half * 2;

    const float bv = bias[o0 + n];
    v8f c;
#pragma unroll
    for (int j = 0; j < 8; ++j) c[j] = bv;

#pragma unroll 4
    for (int i0 = 0; i0 < CIN; i0 += 4) {
        v2f a  = *(const v2f*)(Aptr + i0);
        v2f bb = *(const v2f*)(Bptr + i0);
        c = __builtin_amdgcn_wmma_f32_16x16x4_f32(false, a, false, bb, (short)0, c, false, false);
    }

#pragma unroll
    for (int j = 0; j < 8; ++j) {
        const int m = j + half * 8;
        int tc = chosen[b * KK + m];
        tc = tc < 0 ? 0 : (tc > MAXTIME - 1 ? MAXTIME - 1 : tc);
        const float te = ttab[(size_t)tc * DD + o0 + n];
        entries[((size_t)b * KK + m) * DD + o0 + n] = c[j] + te;
    }
}

extern "C" void kernel_launch(void* const* d_in, const int* in_sizes, int n_in,
                              void* d_out, int out_size, void* d_ws, size_t ws_size,
                              hipStream_t stream) {
    (void)in_sizes; (void)n_in; (void)out_size; (void)ws_size;
    const float* state  = (const float*)d_in[0];   // [32,4096,1024]
    const float* hlog   = (const float*)d_in[1];   // [32,4096,128]
    const float* scores = (const float*)d_in[2];   // [32,4096]
    const float* W      = (const float*)d_in[3];   // [1024,1152]
    const float* bias   = (const float*)d_in[4];   // [1024]
    const float* ttab   = (const float*)d_in[5];   // [512,1024]

    float* out          = (float*)d_out;
    float* out_entries  = out;                                   // 32*16*1024 = 524288
    float* out_valid    = out + (size_t)BB * KK * DD;            // 512
    float* out_times    = out_valid + (size_t)BB * KK;           // 512
    float* out_holder   = out_times + (size_t)BB * KK;           // 32*16*128 = 65536

    int*   chosen = (int*)d_ws;                                  // 2 KB
    float* raw    = (float*)((char*)d_ws + 4096);                // 32*16*1152 f32 = 2.25 MB

    sel_kernel<<<BB, 256, 0, stream>>>(scores, chosen, out_valid, out_times);
    build_raw_kernel<<<dim3(KK, BB), 256, 0, stream>>>(state, hlog, chosen, raw, out_holder);
    gemm_wmma_kernel<<<dim3(DD / (16 * 8), BB), 256, 0, stream>>>(raw, W, bias, ttab, chosen, out_entries);
}